// FlexiHashEmbedding_36378372997500
// MI455X (gfx1250) — compile-verified
//
#include <hip/hip_runtime.h>

#ifndef __has_builtin
#define __has_builtin(x) 0
#endif

#define DIM 128
#define CHUNK 8                 // rows per TDM gather descriptor
#define ITEMS_PER_WAVE 64
#define WAVES_PER_BLOCK 8
#define BLOCK_THREADS (WAVES_PER_BLOCK * 32)

#define AS1 __attribute__((address_space(1)))
#define AS3 __attribute__((address_space(3)))

typedef int v4i __attribute__((vector_size(16)));
typedef unsigned int u32x4 __attribute__((vector_size(16)));
typedef int i32x4 __attribute__((vector_size(16)));
typedef int i32x8 __attribute__((vector_size(32)));

#if __has_builtin(__builtin_amdgcn_tensor_load_to_lds)
#define PATH_TDM 1
#elif __has_builtin(__builtin_amdgcn_global_load_async_to_lds_b128)
#define PATH_ASYNC 1
#else
#define PATH_DIRECT 1
#endif

__device__ __forceinline__ int rfl(int x) {
  return __builtin_amdgcn_readfirstlane(x);
}
__device__ __forceinline__ void compiler_fence() {
  asm volatile("" ::: "memory");
}

// ---------------------------------------------------------------------------
// Zero-init the output (d_out is poisoned by the harness).
// ---------------------------------------------------------------------------
__global__ void zero_out_kernel(float* __restrict__ out, int n4) {
  int i = blockIdx.x * blockDim.x + threadIdx.x;
  if (i < n4) {
    reinterpret_cast<float4*>(out)[i] = make_float4(0.f, 0.f, 0.f, 0.f);
  }
}

// Flush one accumulated segment. Interior segments of a wave's range are
// exclusively owned (embed_index is sorted) -> plain store. Boundary
// segments may be shared with neighboring waves -> atomic f32 adds.
__device__ __forceinline__ void flush_seg(float* __restrict__ out, int seg,
                                          int lane, const float4& acc,
                                          bool use_atomic) {
  float* dst = out + (size_t)seg * DIM + lane * 4;
  if (use_atomic) {
    atomicAdd(dst + 0, acc.x);
    atomicAdd(dst + 1, acc.y);
    atomicAdd(dst + 2, acc.z);
    atomicAdd(dst + 3, acc.w);
  } else {
    *reinterpret_cast<float4*>(dst) = acc;
  }
}

#ifdef PATH_TDM
// ===========================================================================
// Tensor Data Mover path: gather CHUNK=8 embedding rows per descriptor
// (gather mode, 32-bit row indices) into a per-wave double-buffered LDS
// stage; TENSORcnt-pipelined.
// ===========================================================================
__device__ __forceinline__ void wait_tensor_le1() {
#if __has_builtin(__builtin_amdgcn_s_wait_tensorcnt)
  __builtin_amdgcn_s_wait_tensorcnt(1);
#else
  asm volatile("s_wait_tensorcnt 0x1" ::: "memory");
#endif
  compiler_fence();
}
__device__ __forceinline__ void wait_tensor_le0() {
#if __has_builtin(__builtin_amdgcn_s_wait_tensorcnt)
  __builtin_amdgcn_s_wait_tensorcnt(0);
#else
  asm volatile("s_wait_tensorcnt 0x0" ::: "memory");
#endif
  compiler_fence();
}

// Build a gather-mode D# and issue TENSOR_LOAD_TO_LDS for up to 8 rows of
// the [table_rows x DIM] f32 table, starting LDS write at lds_addr.
__device__ __forceinline__ void tdm_issue_gather(
    const float* wbase, unsigned lds_addr, const int* __restrict__ fi, int j,
    int nrows, int table_rows) {
  unsigned long long ga = (unsigned long long)(size_t)wbase;
  u32x4 g0;
  // count=1 | gather_index_size=32b (bit30) | gather_mode=1 (bit31)
  g0[0] = 0xC0000001u;
  g0[1] = lds_addr;                      // LDS byte address
  g0[2] = (unsigned)ga;                  // global_addr[31:0]
  g0[3] = ((unsigned)(ga >> 32) & 0x01FFFFFFu) | (2u << 30);  // addr hi,type=2
  i32x8 g1;
  g1[0] = (int)(2u << 16);               // workgroup_mask=0, data_size=4B
  g1[1] = (int)(((unsigned)DIM & 0xFFFFu) << 16);   // tensor_dim0 lo16
  g1[2] = (int)(((unsigned)table_rows & 0xFFFFu) << 16);  // dim0 hi16=0|dim1 lo
  g1[3] = (int)((((unsigned)table_rows >> 16) & 0xFFFFu) |
                (((unsigned)DIM & 0xFFFFu) << 16));  // dim1 hi16 | tile_dim0
  g1[4] = nrows & 0xFFFF;                // tile_dim1 = #valid indices
  g1[5] = DIM;                           // tensor_dim0_stride[31:0] (elements)
  g1[6] = 0;                             // stride hi | dim1_stride lo (ignored)
  g1[7] = 0;
  i32x4 g2, g3;
#pragma unroll
  for (int r = 0; r < 4; ++r) g2[r] = fi[j + min(r, nrows - 1)];
#pragma unroll
  for (int r = 0; r < 4; ++r) g3[r] = fi[j + min(4 + r, nrows - 1)];
  i32x8 g4 = {0, 0, 0, 0, 0, 0, 0, 0};   // unused 5th SGPR-group operand
  __builtin_amdgcn_tensor_load_to_lds(g0, g1, g2, g3, g4, 0);
}

__global__ __launch_bounds__(BLOCK_THREADS) void gather_scale_segsum_tdm(
    const float* __restrict__ weight, const int* __restrict__ feature_index,
    const int* __restrict__ embed_index, const float* __restrict__ scaling,
    float* __restrict__ out, int nnz, int table_rows) {
  __shared__ __align__(16) float stage[WAVES_PER_BLOCK * 2 * CHUNK * DIM];
  const int lane = (int)(threadIdx.x & 31u);
  const int wib = rfl((int)(threadIdx.x >> 5));
  const int gwave = (int)blockIdx.x * WAVES_PER_BLOCK + wib;
  const int i0 = gwave * ITEMS_PER_WAVE;
  if (i0 >= nnz) return;
  const int iend = min(i0 + ITEMS_PER_WAVE, nnz);
  const int nchunks = (iend - i0 + CHUNK - 1) / CHUNK;

  float* buf0 = &stage[wib * (2 * CHUNK * DIM)];
  float* buf1 = buf0 + CHUNK * DIM;
  const unsigned lds0 = (unsigned)(size_t)(AS3 float*)buf0;
  const unsigned lds1 = (unsigned)(size_t)(AS3 float*)buf1;

  float4 acc = make_float4(0.f, 0.f, 0.f, 0.f);
  int cur_seg = embed_index[rfl(i0)];
  bool first = true;

  // Prologue: stage chunk 0.
  tdm_issue_gather(weight, lds0, feature_index, rfl(i0),
                   min(CHUNK, iend - i0), table_rows);

  for (int c = 0; c < nchunks; ++c) {
    const int base = i0 + c * CHUNK;
    const int n = min(CHUNK, iend - base);
    const bool has_next = (c + 1 < nchunks);
    if (has_next) {
      const int nb = rfl(base + CHUNK);
      tdm_issue_gather(weight, ((c + 1) & 1) ? lds1 : lds0, feature_index, nb,
                       min(CHUNK, iend - nb), table_rows);
    }
    // TDM ops from one wave complete in order: cnt<=1 => chunk c has landed.
    if (has_next) {
      wait_tensor_le1();
    } else {
      wait_tensor_le0();
    }

    const float* buf = (c & 1) ? buf1 : buf0;
    for (int r = 0; r < n; ++r) {
      const int j = rfl(base + r);
      const int seg = embed_index[j];  // wave-uniform -> SMEM
      const float scl = scaling[j];
      const float4 v =
          *reinterpret_cast<const float4*>(buf + r * DIM + lane * 4);
      if (seg != cur_seg) {
        flush_seg(out, cur_seg, lane, acc, first);
        first = false;
        cur_seg = seg;
        acc = make_float4(0.f, 0.f, 0.f, 0.f);
      }
      acc.x = fmaf(scl, v.x, acc.x);
      acc.y = fmaf(scl, v.y, acc.y);
      acc.z = fmaf(scl, v.z, acc.z);
      acc.w = fmaf(scl, v.w, acc.w);
    }
    compiler_fence();  // keep LDS reads before next buffer overwrite
  }
  flush_seg(out, cur_seg, lane, acc, true);
}
#endif  // PATH_TDM

#ifdef PATH_ASYNC
// ===========================================================================
// Fallback: per-row async global->LDS double buffering.
// ===========================================================================
__device__ __forceinline__ void wait_async_le1() {
#if __has_builtin(__builtin_amdgcn_s_wait_asynccnt)
  __builtin_amdgcn_s_wait_asynccnt(1);
#else
  asm volatile("s_wait_asynccnt 0x1" ::: "memory");
#endif
  compiler_fence();
}
__device__ __forceinline__ void wait_async_le0() {
#if __has_builtin(__builtin_amdgcn_s_wait_asynccnt)
  __builtin_amdgcn_s_wait_asynccnt(0);
#else
  asm volatile("s_wait_asynccnt 0x0" ::: "memory");
#endif
  compiler_fence();
}

__global__ __launch_bounds__(BLOCK_THREADS) void gather_scale_segsum_async(
    const float* __restrict__ weight, const int* __restrict__ feature_index,
    const int* __restrict__ embed_index, const float* __restrict__ scaling,
    float* __restrict__ out, int nnz) {
  __shared__ __align__(16) float stage[WAVES_PER_BLOCK * 2 * DIM];
  const int lane = (int)(threadIdx.x & 31u);
  const int wib = rfl((int)(threadIdx.x >> 5));
  const int gwave = (int)blockIdx.x * WAVES_PER_BLOCK + wib;
  const int i0 = gwave * ITEMS_PER_WAVE;
  if (i0 >= nnz) return;
  const int iend = min(i0 + ITEMS_PER_WAVE, nnz);

  float4 acc = make_float4(0.f, 0.f, 0.f, 0.f);
  const int j0 = rfl(i0);
  int fidx = feature_index[j0];
  int seg = embed_index[j0];
  float scl = scaling[j0];
  int cur_seg = seg;
  bool first = true;

  float* mybuf = &stage[wib * (2 * DIM)];
  {
    const float* src = weight + (size_t)fidx * DIM + lane * 4;
    float* dst = mybuf + (i0 & 1) * DIM + lane * 4;
    __builtin_amdgcn_global_load_async_to_lds_b128((AS1 v4i*)src,
                                                   (AS3 v4i*)dst, 0, 0);
  }

  for (int i = i0; i < iend; ++i) {
    const bool has_next = (i + 1 < iend);
    int nfidx = 0, nseg = 0;
    float nscl = 0.f;
    if (has_next) {
      const int j = rfl(i + 1);
      nfidx = feature_index[j];
      nseg = embed_index[j];
      nscl = scaling[j];
      const float* nsrc = weight + (size_t)nfidx * DIM + lane * 4;
      float* ndst = mybuf + ((i + 1) & 1) * DIM + lane * 4;
      __builtin_amdgcn_global_load_async_to_lds_b128((AS1 v4i*)nsrc,
                                                     (AS3 v4i*)ndst, 0, 0);
    }
    if (has_next) {
      wait_async_le1();
    } else {
      wait_async_le0();
    }
    const float4 v =
        *reinterpret_cast<const float4*>(mybuf + (i & 1) * DIM + lane * 4);

    if (seg != cur_seg) {
      flush_seg(out, cur_seg, lane, acc, first);
      first = false;
      cur_seg = seg;
      acc = make_float4(0.f, 0.f, 0.f, 0.f);
    }
    acc.x = fmaf(scl, v.x, acc.x);
    acc.y = fmaf(scl, v.y, acc.y);
    acc.z = fmaf(scl, v.z, acc.z);
    acc.w = fmaf(scl, v.w, acc.w);

    fidx = nfidx;
    seg = nseg;
    scl = nscl;
  }
  flush_seg(out, cur_seg, lane, acc, true);
}
#endif  // PATH_ASYNC

#ifdef PATH_DIRECT
__global__ __launch_bounds__(BLOCK_THREADS) void gather_scale_segsum_direct(
    const float* __restrict__ weight, const int* __restrict__ feature_index,
    const int* __restrict__ embed_index, const float* __restrict__ scaling,
    float* __restrict__ out, int nnz) {
  const int lane = (int)(threadIdx.x & 31u);
  const int wib = rfl((int)(threadIdx.x >> 5));
  const int gwave = (int)blockIdx.x * WAVES_PER_BLOCK + wib;
  const int i0 = gwave * ITEMS_PER_WAVE;
  if (i0 >= nnz) return;
  const int iend = min(i0 + ITEMS_PER_WAVE, nnz);

  float4 acc = make_float4(0.f, 0.f, 0.f, 0.f);
  int cur_seg = embed_index[rfl(i0)];
  bool first = true;

  for (int i = i0; i < iend; ++i) {
    const int j = rfl(i);
    const int seg = embed_index[j];
    const int fidx = feature_index[j];
    const float scl = scaling[j];
    const float4 v = *reinterpret_cast<const float4*>(
        weight + (size_t)fidx * DIM + lane * 4);
    if (seg != cur_seg) {
      flush_seg(out, cur_seg, lane, acc, first);
      first = false;
      cur_seg = seg;
      acc = make_float4(0.f, 0.f, 0.f, 0.f);
    }
    acc.x = fmaf(scl, v.x, acc.x);
    acc.y = fmaf(scl, v.y, acc.y);
    acc.z = fmaf(scl, v.z, acc.z);
    acc.w = fmaf(scl, v.w, acc.w);
  }
  flush_seg(out, cur_seg, lane, acc, true);
}
#endif  // PATH_DIRECT

extern "C" void kernel_launch(void* const* d_in, const int* in_sizes, int n_in,
                              void* d_out, int out_size, void* d_ws,
                              size_t ws_size, hipStream_t stream) {
  const float* weight = (const float*)d_in[0];
  const int* feature_index = (const int*)d_in[1];
  const int* embed_index = (const int*)d_in[2];
  const float* scaling = (const float*)d_in[3];
  float* out = (float*)d_out;

  const int nnz = in_sizes[1];
  const int table_rows = in_sizes[0] / DIM;
  (void)table_rows;

  const int n4 = out_size / 4;
  zero_out_kernel<<<(n4 + 255) / 256, 256, 0, stream>>>(out, n4);

  const int total_waves = (nnz + ITEMS_PER_WAVE - 1) / ITEMS_PER_WAVE;
  const int blocks = (total_waves + WAVES_PER_BLOCK - 1) / WAVES_PER_BLOCK;
#if defined(PATH_TDM)
  gather_scale_segsum_tdm<<<blocks, BLOCK_THREADS, 0, stream>>>(
      weight, feature_index, embed_index, scaling, out, nnz, table_rows);
#elif defined(PATH_ASYNC)
  gather_scale_segsum_async<<<blocks, BLOCK_THREADS, 0, stream>>>(
      weight, feature_index, embed_index, scaling, out, nnz);
#else
  gather_scale_segsum_direct<<<blocks, BLOCK_THREADS, 0, stream>>>(
      weight, feature_index, embed_index, scaling, out, nnz);
#endif
}